// GCNNet_23914377904836
// MI455X (gfx1250) — compile-verified
//
#include <hip/hip_runtime.h>

#define NODES     50000
#define NODES_PAD 50016   // multiple of 32 for 32-row GEMM strips
#define NEDGES    800000
#define NGRAPHS   1000
#define F0 85
#define F1 85
#define F2 170
#define F3 340
#define LD1 96    // pad(85)  -> multiple of 32
#define LD2 192   // pad(170) -> multiple of 32
#define LD3 352   // pad(340) -> multiple of 32

typedef __attribute__((ext_vector_type(2))) float v2f;
typedef __attribute__((ext_vector_type(8))) float v8f;

static inline unsigned cdiv(unsigned a, unsigned b) { return (a + b - 1) / b; }

// ---------------- utility kernels ----------------

__global__ void fill_zero(float* __restrict__ p, unsigned n) {
  unsigned i = blockIdx.x * blockDim.x + threadIdx.x;
  if (i < n) p[i] = 0.0f;
}

// dst[* x pcols] <- src[rows x cols], zero padded, row-major
__global__ void pad_matrix(const float* __restrict__ src, float* __restrict__ dst,
                           unsigned rows, unsigned cols, unsigned pcols, unsigned total) {
  unsigned i = blockIdx.x * blockDim.x + threadIdx.x;
  if (i >= total) return;
  unsigned r = i / pcols, c = i - r * pcols;
  dst[i] = (r < rows && c < cols) ? src[r * cols + c] : 0.0f;
}

// Pack weight W[K x N] (row-major) into WMMA-friendly [Kp/4][Np][4] layout, zero padded:
//   dst[(k>>2)*Np*4 + n*4 + (k&3)] = W[k][n]
// so B[k..k+3][n] are 4 contiguous floats -> lanes do b64 loads of {B[k+kh][n], B[k+kh+1][n]}.
__global__ void pack_weight(const float* __restrict__ src, float* __restrict__ dst,
                            unsigned K, unsigned N, unsigned Np, unsigned total) {
  unsigned i = blockIdx.x * blockDim.x + threadIdx.x;
  if (i >= total) return;
  unsigned k4  = i / (Np * 4);
  unsigned rem = i - k4 * (Np * 4);
  unsigned n   = rem >> 2;
  unsigned k   = (k4 << 2) + (rem & 3);
  dst[i] = (k < K && n < N) ? src[k * N + n] : 0.0f;
}

// ---------------- graph normalization ----------------

__global__ void accum_deg(const int* __restrict__ dst, float* __restrict__ deg, unsigned E) {
  unsigned e = blockIdx.x * blockDim.x + threadIdx.x;
  if (e < E) atomicAdd(&deg[dst[e]], 1.0f);
}

__global__ void compute_dinv(const float* __restrict__ deg, float* __restrict__ dinv, unsigned n) {
  unsigned i = blockIdx.x * blockDim.x + threadIdx.x;
  if (i < n) dinv[i] = rsqrtf(deg[i] + 1.0f);   // +1 = self loop; always > 0
}

__global__ void compute_norm(const int* __restrict__ src, const int* __restrict__ dst,
                             const float* __restrict__ dinv, float* __restrict__ norm, unsigned E) {
  unsigned e = blockIdx.x * blockDim.x + threadIdx.x;
  if (e < E) norm[e] = dinv[src[e]] * dinv[dst[e]];
}

// ---------------- WMMA f32 GEMM: C[M x N] = A[M x K] * Bpacked ----------------
// One wave computes a 32x32 tile: 2 M-tiles x 2 N-tiles = 4 WMMA accumulators.
// Per k-step: 2 A b64 loads + 2 B b64 loads feed 4 v_wmma_f32_16x16x4_f32.
// Requires: M % 32 == 0, N % 32 == 0, K % 4 == 0. lda == K. B in [K/4][N][4] packed layout.

__global__ __launch_bounds__(32) void gemm_wmma_f32(const float* __restrict__ A,
                                                    const float* __restrict__ B,
                                                    float* __restrict__ C,
                                                    int K, int N) {
  const int lane = threadIdx.x & 31;
  const int m0 = blockIdx.x << 5;    // 32 rows per wave
  const int n0 = blockIdx.y << 5;    // 32 cols per wave
  const int r  = lane & 15;          // A row in tile / B,C column in tile
  const int kh = (lane >> 4) << 1;   // lanes 0-15 -> K0,K1 ; lanes 16-31 -> K2,K3

  const float* A0 = A + (size_t)(m0 + r) * K + kh;
  const float* A1 = A0 + (size_t)16 * K;
  const float* Bb = B + (size_t)(n0 + r) * 4 + kh;   // + (k>>2)*N*4 per k-step
  const size_t bstep = (size_t)N * 4;                // floats per 4-row B slab

  v8f c00 = {0.f,0.f,0.f,0.f,0.f,0.f,0.f,0.f};
  v8f c01 = c00, c10 = c00, c11 = c00;

  for (int k = 0; k < K; k += 4) {
    const v2f a0 = *(const v2f*)(A0 + k);
    const v2f a1 = *(const v2f*)(A1 + k);
    const float* Bp = Bb + (size_t)(k >> 2) * bstep;
    const v2f b0 = *(const v2f*)(Bp);
    const v2f b1 = *(const v2f*)(Bp + 64);           // +16 columns * 4
    c00 = __builtin_amdgcn_wmma_f32_16x16x4_f32(false, a0, false, b0, (short)0, c00, false, false);
    c01 = __builtin_amdgcn_wmma_f32_16x16x4_f32(false, a0, false, b1, (short)0, c01, false, false);
    c10 = __builtin_amdgcn_wmma_f32_16x16x4_f32(false, a1, false, b0, (short)0, c10, false, false);
    c11 = __builtin_amdgcn_wmma_f32_16x16x4_f32(false, a1, false, b1, (short)0, c11, false, false);
  }

  // C/D layout: VGPR v -> row v + 8*(lane>>4), col = lane&15
  float* Cp0 = C + (size_t)(m0 + ((lane >> 4) << 3)) * N + n0 + r;
  float* Cp1 = Cp0 + (size_t)16 * N;
  #pragma unroll
  for (int v = 0; v < 8; ++v) {
    Cp0[(size_t)v * N]      = c00[v];
    Cp0[(size_t)v * N + 16] = c01[v];
    Cp1[(size_t)v * N]      = c10[v];
    Cp1[(size_t)v * N + 16] = c11[v];
  }
}

// ---------------- edge aggregation: out[dst] += norm * h[src] ----------------

__global__ void aggregate(const float* __restrict__ h, const int* __restrict__ src,
                          const int* __restrict__ dst, const float* __restrict__ norm,
                          float* __restrict__ out, unsigned E, int ld) {
  unsigned tid = blockIdx.x * blockDim.x + threadIdx.x;
  const unsigned chunks = (unsigned)(ld >> 2);
  const unsigned total = E * chunks;
  if (tid >= total) return;
  unsigned e = tid / chunks;
  unsigned c = (tid - e * chunks) << 2;
  const int s = src[e], d = dst[e];
  const float w = norm[e];
  const float4 hv = *(const float4*)(h + (size_t)s * ld + c);
  float* o = out + (size_t)d * ld + c;
  atomicAdd(o + 0, hv.x * w);
  atomicAdd(o + 1, hv.y * w);
  atomicAdd(o + 2, hv.z * w);
  atomicAdd(o + 3, hv.w * w);
}

// ---------------- self-loop + bias + relu (in place on agg buffer) ----------------

__global__ void bias_relu_self(float* __restrict__ out, const float* __restrict__ h,
                               const float* __restrict__ dinv, const float* __restrict__ bias,
                               unsigned n, int ld, int dvalid) {
  unsigned tid = blockIdx.x * blockDim.x + threadIdx.x;
  unsigned total = n * (unsigned)ld;
  if (tid >= total) return;
  unsigned i = tid / (unsigned)ld;
  unsigned j = tid - i * (unsigned)ld;
  float v = 0.0f;
  if ((int)j < dvalid) {
    const float di = dinv[i];
    v = out[tid] + di * di * h[tid] + bias[j];
    v = fmaxf(v, 0.0f);
  }
  out[tid] = v;
}

// ---------------- global max pool (values >= 0 after relu) ----------------

__global__ void pool_max(const float* __restrict__ act, const int* __restrict__ batch,
                         unsigned* __restrict__ pooled, unsigned n, int ld, int dvalid) {
  unsigned tid = blockIdx.x * blockDim.x + threadIdx.x;
  unsigned total = n * (unsigned)ld;
  if (tid >= total) return;
  unsigned i = tid / (unsigned)ld;
  unsigned j = tid - i * (unsigned)ld;
  if ((int)j >= dvalid) return;
  atomicMax(&pooled[(size_t)batch[i] * ld + j], __float_as_uint(act[tid]));
}

// ---------------- fused FC head: relu(pooled@Wfc1+b1)@Wfc2+b2 ----------------

__global__ __launch_bounds__(256) void fc_head(const float* __restrict__ pooled,
                                               const float* __restrict__ Wfc1,
                                               const float* __restrict__ bfc1,
                                               const float* __restrict__ Wfc2,
                                               const float* __restrict__ bfc2,
                                               float* __restrict__ out) {
  __shared__ float srow[LD3];
  __shared__ float red[256];
  const int g = blockIdx.x, t = threadIdx.x;
  for (int k = t; k < LD3; k += 256) srow[k] = pooled[(size_t)g * LD3 + k];
  __syncthreads();
  float partial = 0.0f;
  if (t < 218) {
    float s = 0.0f;
    for (int k = 0; k < F3; ++k) s += srow[k] * Wfc1[k * 218 + t];
    s = fmaxf(s + bfc1[t], 0.0f);
    partial = s * Wfc2[t];
  }
  red[t] = partial;
  __syncthreads();
  for (int o = 128; o > 0; o >>= 1) {
    if (t < o) red[t] += red[t + o];
    __syncthreads();
  }
  if (t == 0) out[g] = red[0] + bfc2[0];
}

// ---------------- launcher ----------------

extern "C" void kernel_launch(void* const* d_in, const int* in_sizes, int n_in,
                              void* d_out, int out_size, void* d_ws, size_t ws_size,
                              hipStream_t stream) {
  const float* x    = (const float*)d_in[0];
  const int*   ei   = (const int*)d_in[1];     // (2, E) flat
  const int*   bat  = (const int*)d_in[2];
  const float* W1   = (const float*)d_in[3];
  const float* b1   = (const float*)d_in[4];
  const float* W2   = (const float*)d_in[5];
  const float* b2   = (const float*)d_in[6];
  const float* W3   = (const float*)d_in[7];
  const float* b3   = (const float*)d_in[8];
  const float* Wfc1 = (const float*)d_in[9];
  const float* bfc1 = (const float*)d_in[10];
  const float* Wfc2 = (const float*)d_in[11];
  const float* bfc2 = (const float*)d_in[12];
  float* out = (float*)d_out;

  const int* src = ei;
  const int* dst = ei + NEDGES;

  // workspace carve-up (floats)
  float* ws = (float*)d_ws;
  size_t o = 0;
  float* Xp   = ws + o; o += (size_t)NODES_PAD * LD1;
  float* W1p  = ws + o; o += (size_t)LD1 * LD1;
  float* W2p  = ws + o; o += (size_t)LD1 * LD2;
  float* W3p  = ws + o; o += (size_t)LD2 * LD3;
  float* deg  = ws + o; o += NODES;
  float* dinv = ws + o; o += NODES;
  float* norm = ws + o; o += NEDGES;
  float* hbuf = ws + o; o += (size_t)NODES_PAD * LD3;
  float* act1 = ws + o; o += (size_t)NODES_PAD * LD1;
  float* act2 = ws + o; o += (size_t)NODES_PAD * LD2;
  float* act3 = ws + o; o += (size_t)NODES_PAD * LD3;
  float* pooled = ws + o; o += (size_t)NGRAPHS * LD3;
  (void)ws_size; (void)n_in; (void)in_sizes; (void)out_size;

  const int TB = 256;

  // ---- pad input, pack weights into [K/4][N][4] ----
  { unsigned t = NODES_PAD * LD1;
    pad_matrix<<<cdiv(t, TB), TB, 0, stream>>>(x, Xp, NODES, F0, LD1, t); }
  { unsigned t = LD1 * LD1;
    pack_weight<<<cdiv(t, TB), TB, 0, stream>>>(W1, W1p, F1, F1, LD1, t); }
  { unsigned t = LD1 * LD2;
    pack_weight<<<cdiv(t, TB), TB, 0, stream>>>(W2, W2p, F1, F2, LD2, t); }
  { unsigned t = LD2 * LD3;
    pack_weight<<<cdiv(t, TB), TB, 0, stream>>>(W3, W3p, F2, F3, LD3, t); }

  // ---- symmetric normalization ----
  fill_zero<<<cdiv(NODES, TB), TB, 0, stream>>>(deg, NODES);
  accum_deg<<<cdiv(NEDGES, TB), TB, 0, stream>>>(dst, deg, NEDGES);
  compute_dinv<<<cdiv(NODES, TB), TB, 0, stream>>>(deg, dinv, NODES);
  compute_norm<<<cdiv(NEDGES, TB), TB, 0, stream>>>(src, dst, dinv, norm, NEDGES);

  const unsigned MT = NODES_PAD / 32;  // 1563 row strips

  // ---- layer 1: 96 -> 96 ----
  gemm_wmma_f32<<<dim3(MT, LD1 / 32), 32, 0, stream>>>(Xp, W1p, hbuf, LD1, LD1);
  { unsigned tp = NODES_PAD * LD1, t = NODES * LD1;
    fill_zero<<<cdiv(tp, TB), TB, 0, stream>>>(act1, tp);
    aggregate<<<cdiv(NEDGES * (LD1 / 4), TB), TB, 0, stream>>>(hbuf, src, dst, norm, act1, NEDGES, LD1);
    bias_relu_self<<<cdiv(t, TB), TB, 0, stream>>>(act1, hbuf, dinv, b1, NODES, LD1, F1); }

  // ---- layer 2: 96 -> 192 ----
  gemm_wmma_f32<<<dim3(MT, LD2 / 32), 32, 0, stream>>>(act1, W2p, hbuf, LD1, LD2);
  { unsigned tp = NODES_PAD * LD2, t = NODES * LD2;
    fill_zero<<<cdiv(tp, TB), TB, 0, stream>>>(act2, tp);
    aggregate<<<cdiv(NEDGES * (LD2 / 4), TB), TB, 0, stream>>>(hbuf, src, dst, norm, act2, NEDGES, LD2);
    bias_relu_self<<<cdiv(t, TB), TB, 0, stream>>>(act2, hbuf, dinv, b2, NODES, LD2, F2); }

  // ---- layer 3: 192 -> 352 ----
  gemm_wmma_f32<<<dim3(MT, LD3 / 32), 32, 0, stream>>>(act2, W3p, hbuf, LD2, LD3);
  { unsigned tp = NODES_PAD * LD3, t = NODES * LD3;
    fill_zero<<<cdiv(tp, TB), TB, 0, stream>>>(act3, tp);
    aggregate<<<cdiv(NEDGES * (LD3 / 4), TB), TB, 0, stream>>>(hbuf, src, dst, norm, act3, NEDGES, LD3);
    bias_relu_self<<<cdiv(t, TB), TB, 0, stream>>>(act3, hbuf, dinv, b3, NODES, LD3, F3); }

  // ---- global max pool ----
  { unsigned t = NGRAPHS * LD3;
    fill_zero<<<cdiv(t, TB), TB, 0, stream>>>(pooled, t); }
  { unsigned t = NODES * LD3;
    pool_max<<<cdiv(t, TB), TB, 0, stream>>>(act3, bat, (unsigned*)pooled, NODES, LD3, F3); }

  // ---- MLP head ----
  fc_head<<<NGRAPHS, 256, 0, stream>>>(pooled, Wfc1, bfc1, Wfc2, bfc2, out);
}